// GraphConv_65481071394920
// MI455X (gfx1250) — compile-verified
//
#include <hip/hip_runtime.h>

#define NN 100000
#define NE 1600000
#define DIM 32

typedef __attribute__((ext_vector_type(2))) float v2f;
typedef __attribute__((ext_vector_type(8))) float v8f;

// ---------------- zero scratch (deg_out, deg_in, agg) ----------------
__global__ void zero_kernel(float* __restrict__ p, int n) {
    int i = blockIdx.x * blockDim.x + threadIdx.x;
    if (i < n) p[i] = 0.0f;
}

// ---------------- degree counts via float atomics (exact up to 2^24) --
__global__ void degree_kernel(const int* __restrict__ src, const int* __restrict__ dst,
                              float* __restrict__ deg_out, float* __restrict__ deg_in, int E) {
    int e = blockIdx.x * blockDim.x + threadIdx.x;
    if (e < E) {
        atomicAdd(&deg_out[src[e]], 1.0f);
        atomicAdd(&deg_in[dst[e]], 1.0f);
    }
}

// ---------------- h = X * rsqrt(max(deg_out,1)) -----------------------
__global__ void scale_kernel(const float* __restrict__ feat, const float* __restrict__ deg_out,
                             float* __restrict__ h, int n) {
    int i = blockIdx.x * blockDim.x + threadIdx.x;   // over N*32
    if (i < n) {
        float d = deg_out[i >> 5];
        h[i] = feat[i] * __frsqrt_rn(fmaxf(d, 1.0f));
    }
}

// ---------------- agg[dst] += h[src]  (32 lanes per edge) -------------
// Consecutive lanes cover the 32 features of one edge: coalesced gather of
// h[src] and contiguous L2 atomics into one agg row. All node arrays are
// L2-resident (12.8 MB each vs 192 MB L2).
__global__ void scatter_kernel(const int* __restrict__ src, const int* __restrict__ dst,
                               const float* __restrict__ h, float* __restrict__ agg, int E) {
    int gid = blockIdx.x * blockDim.x + threadIdx.x;  // E*32 = 51.2M < 2^31
    int e = gid >> 5;
    int d = gid & 31;
    if (e < E) {
        atomicAdd(&agg[dst[e] * DIM + d], h[src[e] * DIM + d]);
    }
}

// ---------------- out = relu((agg * rsqrt(deg_in)) @ W + b) -----------
// One wave computes a 16-row x 32-col output tile using V_WMMA_F32_16X16X4_F32.
// A (16x4 f32): lanes 0-15 hold M=lane, K={0,1}; lanes 16-31 hold M=lane-16, K={2,3}.
// B (4x16 f32): mirrored striping, N = lane&15.
// C/D (16x16 f32): VGPR r holds M = r + (lane<16 ? 0 : 8), N = lane&15.
__global__ void gemm_relu_kernel(const float* __restrict__ agg, const float* __restrict__ deg_in,
                                 const float* __restrict__ W, const float* __restrict__ bias,
                                 float* __restrict__ out, int numTiles) {
    const int lane = threadIdx.x & 31;
    const int wave = threadIdx.x >> 5;
    const int tile = blockIdx.x * (blockDim.x >> 5) + wave;
    if (tile >= numTiles) return;              // wave-uniform: EXEC all-ones inside

    const int m    = lane & 15;
    const int rowA = tile * 16 + m;            // A row this lane feeds
    const float nrm = __frsqrt_rn(fmaxf(deg_in[rowA], 1.0f));
    const int koff = (lane < 16) ? 0 : 2;      // K sub-offset per ISA layout

    v8f c0 = {};   // output cols 0..15
    v8f c1 = {};   // output cols 16..31

#pragma unroll
    for (int kb = 0; kb < DIM; kb += 4) {
        const int k0 = kb + koff;
        v2f a;
        a.x = agg[rowA * DIM + k0]     * nrm;
        a.y = agg[rowA * DIM + k0 + 1] * nrm;

        v2f b0, b1;
        b0.x = W[(k0)     * DIM + m];
        b0.y = W[(k0 + 1) * DIM + m];
        b1.x = W[(k0)     * DIM + 16 + m];
        b1.y = W[(k0 + 1) * DIM + 16 + m];

        c0 = __builtin_amdgcn_wmma_f32_16x16x4_f32(false, a, false, b0, (short)0, c0, false, false);
        c1 = __builtin_amdgcn_wmma_f32_16x16x4_f32(false, a, false, b1, (short)0, c1, false, false);
    }

    const int rbase = tile * 16 + ((lane < 16) ? 0 : 8);
    const float bia0 = bias[m];
    const float bia1 = bias[16 + m];
#pragma unroll
    for (int r = 0; r < 8; ++r) {
        const int row = rbase + r;
        out[row * DIM + m]      = fmaxf(c0[r] + bia0, 0.0f);
        out[row * DIM + 16 + m] = fmaxf(c1[r] + bia1, 0.0f);
    }
}

extern "C" void kernel_launch(void* const* d_in, const int* in_sizes, int n_in,
                              void* d_out, int out_size, void* d_ws, size_t ws_size,
                              hipStream_t stream) {
    const float* features = (const float*)d_in[0];
    const int*   src      = (const int*)d_in[1];
    const int*   dst      = (const int*)d_in[2];
    const float* weight   = (const float*)d_in[3];
    const float* bias     = (const float*)d_in[4];
    float*       out      = (float*)d_out;

    const int E = in_sizes[1];     // 1,600,000
    const int N = NN;              // 100,000

    // Workspace layout: deg_out[N] | deg_in[N] | agg[N*32] | h[N*32]
    float* deg_out = (float*)d_ws;
    float* deg_in  = deg_out + N;
    float* agg     = deg_in + N;
    float* h       = agg + (size_t)N * DIM;

    // 1) zero deg_out, deg_in, agg (contiguous 34N floats); h fully overwritten later
    const int zn = N * (2 + DIM);
    zero_kernel<<<(zn + 255) / 256, 256, 0, stream>>>((float*)d_ws, zn);

    // 2) degrees
    degree_kernel<<<(E + 255) / 256, 256, 0, stream>>>(src, dst, deg_out, deg_in, E);

    // 3) h = X * rsqrt(max(deg_out,1))
    const int hn = N * DIM;
    scale_kernel<<<(hn + 255) / 256, 256, 0, stream>>>(features, deg_out, h, hn);

    // 4) agg[dst] += h[src], 32 lanes/edge
    const long long sthreads = (long long)E * DIM;
    scatter_kernel<<<(int)((sthreads + 255) / 256), 256, 0, stream>>>(src, dst, h, agg, E);

    // 5) WMMA GEMM + bias + ReLU  (N/16 = 6250 tiles, 8 waves per 256-thread block)
    const int numTiles = N / 16;
    const int wavesPerBlock = 8;
    const int blocks = (numTiles + wavesPerBlock - 1) / wavesPerBlock;
    gemm_relu_kernel<<<blocks, wavesPerBlock * 32, 0, stream>>>(agg, deg_in, weight, bias, out, numTiles);
}